// PAM_31756988187044
// MI455X (gfx1250) — compile-verified
//
#include <hip/hip_runtime.h>
#include <hip/hip_bf16.h>
#include <cstddef>

// Problem constants (from reference): x[2,64,64,64,64], preds[2,8,64,64,64]
#define BSZ   2
#define CSZ   64
#define KSZ   8
#define BKSZ  16            // B*K segments -> exactly WMMA M=16
#define NVOX  (64*64*64)    // 262144 voxels per batch
#define TILE_V 64           // voxels staged per block iteration
#define TILE_PAD 4          // row stride 68 floats = 272B (16B aligned, conflict-free)

typedef float v2f __attribute__((ext_vector_type(2)));
typedef float v8f __attribute__((ext_vector_type(8)));

// ---- monotonic float<->uint encoding for atomic max on floats ----
__device__ __forceinline__ unsigned int enc_f32(float f) {
    unsigned int b = __float_as_uint(f);
    return (b & 0x80000000u) ? ~b : (b | 0x80000000u);
}
__device__ __forceinline__ float dec_f32(unsigned int e) {
    unsigned int b = (e & 0x80000000u) ? (e & 0x7fffffffu) : ~e;
    return __uint_as_float(b);
}

// ============================================================
// Kernel 1: per-voxel argmax over K=8 preds (4 voxels/thread,
// float4 loads -> global_load_b128), store seg (uchar4) and
// winning logits (float4); segment-max via LDS + global atomicMax.
// ============================================================
__global__ void pam_argmax(const float* __restrict__ preds,
                           unsigned char* __restrict__ segb,
                           float* __restrict__ logitb,
                           unsigned int* __restrict__ lmax_enc) {
    __shared__ unsigned int lmax_s[BKSZ];
    const int t = threadIdx.x;
    if (t < BKSZ) lmax_s[t] = 0u;
    __syncthreads();

    const size_t idx4 = (size_t)blockIdx.x * blockDim.x + t;  // over B*N/4 (exact grid)
    const size_t flat = idx4 * 4;
    const int b = (int)(flat / NVOX);
    const int n = (int)(flat % NVOX);
    const float* p = preds + (size_t)b * KSZ * NVOX + n;

    float4 best = *(const float4*)p;
    int ax = 0, ay = 0, az = 0, aw = 0;
#pragma unroll
    for (int k = 1; k < KSZ; ++k) {
        const float4 v = *(const float4*)(p + (size_t)k * NVOX);
        if (v.x > best.x) { best.x = v.x; ax = k; }   // first-max tiebreak
        if (v.y > best.y) { best.y = v.y; ay = k; }
        if (v.z > best.z) { best.z = v.z; az = k; }
        if (v.w > best.w) { best.w = v.w; aw = k; }
    }
    const int base = b * KSZ;                         // fold batch into segment id
    uchar4 sg;
    sg.x = (unsigned char)(base + ax);
    sg.y = (unsigned char)(base + ay);
    sg.z = (unsigned char)(base + az);
    sg.w = (unsigned char)(base + aw);
    *(uchar4*)(segb + flat)   = sg;
    *(float4*)(logitb + flat) = best;

    atomicMax(&lmax_s[base + ax], enc_f32(best.x));
    atomicMax(&lmax_s[base + ay], enc_f32(best.y));
    atomicMax(&lmax_s[base + az], enc_f32(best.z));
    atomicMax(&lmax_s[base + aw], enc_f32(best.w));
    __syncthreads();
    if (t < BKSZ) atomicMax(&lmax_enc[t], lmax_s[t]);
}

// ============================================================
// Kernel 2: WMMA reduction.
//   wfeat[16x64] += A[16xK] * X[Kx64],  A[s,v] = w[v] * (seg[v]==s)
// 256 threads (8 waves). Per tile: stage 64ch x 64vox of x into LDS.
// Staging batches 4x global_load_b128 into registers (4 loads in
// flight, single wait) then 4x ds_store_b128; next tile prefetched.
// Each wave consumes 8 voxels as two V_WMMA_F32_16X16X4_F32 K-steps
// per 16-channel chunk (4 chunks). wsum/cnt fall out of the masked
// A values. Block-level LDS reduction, then global float atomics.
// ============================================================
__global__ void pam_wmma_accum(const float* __restrict__ x,
                               const unsigned char* __restrict__ segb,
                               const float* __restrict__ logitb,
                               const unsigned int* __restrict__ lmax_enc,
                               float* __restrict__ wfeat,
                               float* __restrict__ wsum,
                               float* __restrict__ cnt) {
    __shared__ __attribute__((aligned(16))) float xs[CSZ][TILE_V + TILE_PAD];
    __shared__ float wls[TILE_V];
    __shared__ int   sgs[TILE_V];
    __shared__ float lmax_s[BKSZ];
    __shared__ float wfeat_s[BKSZ * CSZ];
    __shared__ float wsum_s[BKSZ];
    __shared__ float cnt_s[BKSZ];

    const int t = threadIdx.x;
    if (t < BKSZ) {
        lmax_s[t] = dec_f32(lmax_enc[t]);
        wsum_s[t] = 0.f;
        cnt_s[t]  = 0.f;
    }
    for (int i = t; i < BKSZ * CSZ; i += blockDim.x) wfeat_s[i] = 0.f;
    __syncthreads();

    const int lane = t & 31;
    const int wv   = t >> 5;                  // wave id 0..7
    const int m    = lane & 15;               // WMMA M (=segment row) / N (=channel col)
    const int kb   = (lane < 16) ? 0 : 2;     // K-half per A/B VGPR layout

    v8f acc[4] = { {0,0,0,0,0,0,0,0}, {0,0,0,0,0,0,0,0},
                   {0,0,0,0,0,0,0,0}, {0,0,0,0,0,0,0,0} };
    float ws_acc = 0.f, ct_acc = 0.f;

    const int tiles_total     = (BSZ * NVOX) / TILE_V;       // 8192
    const int tiles_per_block = tiles_total / gridDim.x;
    const int tile0           = blockIdx.x * tiles_per_block;

    const int vg = (t & 15) * 4;              // voxel group start 0..60
    const int c0 = t >> 4;                    // channel 0..15

    for (int tt = 0; tt < tiles_per_block; ++tt) {
        const int tile = tile0 + tt;
        const int b    = tile / (NVOX / TILE_V);
        const int n0   = (tile % (NVOX / TILE_V)) * TILE_V;

        // ---- stage x tile: 4 b128 loads in flight, then 4 b128 LDS stores ----
        {
            const float* xb = x + (size_t)b * CSZ * NVOX + n0 + vg;
            const float4 r0 = *(const float4*)(xb + (size_t)(c0     ) * NVOX);
            const float4 r1 = *(const float4*)(xb + (size_t)(c0 + 16) * NVOX);
            const float4 r2 = *(const float4*)(xb + (size_t)(c0 + 32) * NVOX);
            const float4 r3 = *(const float4*)(xb + (size_t)(c0 + 48) * NVOX);
            *(float4*)&xs[c0     ][vg] = r0;
            *(float4*)&xs[c0 + 16][vg] = r1;
            *(float4*)&xs[c0 + 32][vg] = r2;
            *(float4*)&xs[c0 + 48][vg] = r3;
            if (tt + 1 < tiles_per_block) {   // pull next 16KB tile toward cache
                __builtin_prefetch(xb + TILE_V, 0, 1);
                __builtin_prefetch(xb + (size_t)(c0 + 32) * NVOX + TILE_V, 0, 1);
            }
        }
        if (t < TILE_V) {
            const int   s = segb[(size_t)b * NVOX + n0 + t];
            const float l = logitb[(size_t)b * NVOX + n0 + t];
            sgs[t] = s;
            wls[t] = expf(l - lmax_s[s]);     // numerically-stable softmax weight
        }
        __syncthreads();

        // ---- two K=4 WMMA steps per wave (8 waves cover 64 voxels) ----
#pragma unroll
        for (int j = 0; j < 2; ++j) {
            const int k0 = wv * 8 + j * 4 + kb;
            const int k1 = k0 + 1;
            const int s0 = sgs[k0], s1 = sgs[k1];
            const float w0 = wls[k0], w1 = wls[k1];

            v2f a;
            a.x = (s0 == m) ? w0 : 0.f;       // one-hot, softmax-weighted A
            a.y = (s1 == m) ? w1 : 0.f;
            ws_acc += a.x + a.y;              // lanes L and L+16 cover disjoint K
            ct_acc += (float)((s0 == m) + (s1 == m));

#pragma unroll
            for (int ch = 0; ch < 4; ++ch) {
                v2f bv;
                bv.x = xs[ch * 16 + m][k0];   // B[K,N]: row=voxel, col=channel
                bv.y = xs[ch * 16 + m][k1];
                acc[ch] = __builtin_amdgcn_wmma_f32_16x16x4_f32(
                    false, a, false, bv, (short)0, acc[ch], false, false);
            }
        }
        __syncthreads();
    }

    // ---- reduce: C/D layout VGPR r, lanes<16 -> M=r, lanes>=16 -> M=r+8 ----
    const int seg_hi = (lane < 16) ? 0 : 8;
#pragma unroll
    for (int ch = 0; ch < 4; ++ch)
#pragma unroll
        for (int r = 0; r < 8; ++r)
            atomicAdd(&wfeat_s[(r + seg_hi) * CSZ + ch * 16 + m], acc[ch][r]);
    atomicAdd(&wsum_s[m], ws_acc);
    atomicAdd(&cnt_s[m],  ct_acc);
    __syncthreads();

    for (int i = t; i < BKSZ * CSZ; i += blockDim.x) atomicAdd(&wfeat[i], wfeat_s[i]);
    if (t < BKSZ) {
        atomicAdd(&wsum[t], wsum_s[t]);
        atomicAdd(&cnt[t],  cnt_s[t]);
    }
}

// ============================================================
// Kernel 3: per_seg = wfeat / (wsum*cnt), safe for empty segments.
// ============================================================
__global__ void pam_finalize(const float* __restrict__ wfeat,
                             const float* __restrict__ wsum,
                             const float* __restrict__ cnt,
                             float* __restrict__ per_seg) {
    for (int i = threadIdx.x; i < BKSZ * CSZ; i += blockDim.x) {
        const int s = i / CSZ;
        const float denom = (cnt[s] > 0.f) ? (wsum[s] * cnt[s]) : 1.f;
        per_seg[i] = wfeat[i] / denom;
    }
}

// ============================================================
// Kernel 4: out[b,c,n] = x[b,c,n] * per_seg[seg(b,n), c]
// float4 streaming; 4KB per_seg table in LDS; seg as uchar4.
// ============================================================
__global__ void pam_scatter(const float* __restrict__ x,
                            const unsigned char* __restrict__ segb,
                            const float* __restrict__ per_seg,
                            float* __restrict__ out) {
    __shared__ float ps[BKSZ * CSZ];
    for (int i = threadIdx.x; i < BKSZ * CSZ; i += blockDim.x) ps[i] = per_seg[i];
    __syncthreads();

    const size_t idx4 = (size_t)blockIdx.x * blockDim.x + threadIdx.x;  // exact grid
    const size_t flat = idx4 * 4;
    const int    b   = (int)(flat / ((size_t)CSZ * NVOX));
    const size_t rem = flat % ((size_t)CSZ * NVOX);
    const int    c   = (int)(rem / NVOX);
    const int    n   = (int)(rem % NVOX);

    const float4 xv = *(const float4*)(x + flat);
    const uchar4 sg = *(const uchar4*)(segb + (size_t)b * NVOX + n);
    float4 o;
    o.x = xv.x * ps[(int)sg.x * CSZ + c];
    o.y = xv.y * ps[(int)sg.y * CSZ + c];
    o.z = xv.z * ps[(int)sg.z * CSZ + c];
    o.w = xv.w * ps[(int)sg.w * CSZ + c];
    *(float4*)(out + flat) = o;
}

// ============================================================
// Host launcher
// ============================================================
extern "C" void kernel_launch(void* const* d_in, const int* in_sizes, int n_in,
                              void* d_out, int out_size, void* d_ws, size_t ws_size,
                              hipStream_t stream) {
    const float* x     = (const float*)d_in[0];   // [2,64,64,64,64]
    const float* preds = (const float*)d_in[1];   // [2,8,64,64,64]
    float* out = (float*)d_out;

    // workspace layout
    char* ws = (char*)d_ws;
    const size_t SEG_BYTES   = (size_t)BSZ * NVOX;            // 512 KB (u8)
    const size_t LOGIT_BYTES = (size_t)BSZ * NVOX * 4;        // 2 MB
    unsigned char* segb   = (unsigned char*)ws;
    float*         logitb = (float*)(ws + SEG_BYTES);
    char* small = ws + SEG_BYTES + LOGIT_BYTES;
    unsigned int* lmax_enc = (unsigned int*)(small + 0);      // 16 u32
    float*        wsum     = (float*)(small + 64);            // 16 f32
    float*        cnt      = (float*)(small + 128);           // 16 f32
    float*        wfeat    = (float*)(small + 192);           // 1024 f32
    float*        per_seg  = (float*)(small + 192 + 4096);    // 1024 f32
    const size_t SMALL_BYTES = 192 + 4096 + 4096;

    // zero accumulators (lmax enc=0 == "-inf" in monotonic encoding)
    hipMemsetAsync(small, 0, SMALL_BYTES, stream);

    // 1) argmax + logit + segment max      (reads preds 17 MB, b128 loads)
    pam_argmax<<<(BSZ * NVOX / 4) / 256, 256, 0, stream>>>(preds, segb, logitb, lmax_enc);

    // 2) WMMA segment-weighted feature sum (reads x once, 134 MB, b128 staged)
    pam_wmma_accum<<<512, 256, 0, stream>>>(x, segb, logitb, lmax_enc,
                                            wfeat, wsum, cnt);

    // 3) tiny finalize
    pam_finalize<<<1, 256, 0, stream>>>(wfeat, wsum, cnt, per_seg);

    // 4) scatter-multiply                  (reads x + writes out, 268 MB)
    pam_scatter<<<((size_t)BSZ * CSZ * NVOX / 4) / 256, 256, 0, stream>>>(
        x, segb, per_seg, out);
}